// RFSConv_71382356459645
// MI455X (gfx1250) — compile-verified
//
#include <hip/hip_runtime.h>
#include <hip/hip_bf16.h>

// CDNA5 / gfx1250 implementation of the CARAFE-style upsample block.
//
// Key algebraic fusion: the stride-3 conv over the K=3 pixel-shuffled tensor
// collapses to a per-pixel GEMV out[64] = w_conv[64x594] * g[594], so both
// heavy stages are GEMMs mapped onto V_WMMA_F32_16X16X4_F32 (fp32 matrix op,
// bit-compatible precision with the fp32 reference).
//
// gfx1250-specific paths used:
//   * v_wmma_f32_16x16x4_f32 for both GEMMs (fp32 in/out)
//   * global_load_async_to_lds_b32 (ASYNCcnt DMA) for the K1 input staging
//
// d_ws layout: up-buffer (squeeze output, top 64 channels):
//   float up[8][64][128][128]  -> 33,554,432 bytes required.

typedef __attribute__((ext_vector_type(2))) float v2f;
typedef __attribute__((ext_vector_type(8))) float v8f;

#define HW 128            // spatial size after unshuffle
#define B_ 8
#define C_SQ_IN 256       // 4*64 unshuffled channels
#define C_SQ_OUT 128
#define CIN_ 66           // 64 + 2 coord channels
#define CK 594            // CIN_*9
#define CKPAD 596         // padded to multiple of 4 for WMMA K loop

// final channel shuffle: concat-channel k (0..127) -> output channel
__device__ __forceinline__ int shuffle_ch(int k) {
    return (k & 1) ? (64 + (k >> 1)) : (k >> 1);
}

// Async DMA of one b32 element from global memory straight into LDS.
// LDS byte address comes from the VDST VGPR, global address from the VADDR
// pair (GV mode, SADDR=off). Tracked by ASYNCcnt.
__device__ __forceinline__ void async_g2lds_b32(unsigned lds_off,
                                                unsigned long long gaddr) {
    asm volatile("global_load_async_to_lds_b32 %0, %1, off"
                 :
                 : "v"(lds_off), "v"(gaddr)
                 : "memory");
}

__device__ __forceinline__ void wait_asynccnt0() {
    asm volatile("s_wait_asynccnt 0x0" ::: "memory");
}

// ---------------------------------------------------------------------------
// Kernel 1: pixel-unshuffle + 1x1 squeeze conv as WMMA GEMM.
//   M = 128 out channels (8 waves x 16), K = 256, N = 16 pixels per block.
//   m <  64 -> up buffer (workspace), m >= 64 -> "low" half written straight
//   to d_out at the channel-shuffled position.
// ---------------------------------------------------------------------------
__global__ __launch_bounds__(256) void k_squeeze(
    const float* __restrict__ x,      // (8,64,256,256)
    const float* __restrict__ wsq,    // (128,256)
    float* __restrict__ up,           // (8,64,128,128) workspace
    float* __restrict__ out)          // (8,128,128,128)
{
    __shared__ float ldsX[C_SQ_IN * 16];   // xin[k][n], 16 KB

    const int p0  = blockIdx.x * 16;       // first pixel of this N-tile
    const int bi  = p0 >> 14;              // / (128*128)
    const int rem = p0 & 16383;
    const int y   = rem >> 7;
    const int x0  = rem & 127;             // 16-aligned
    const int tid = threadIdx.x;

    // Stage the 256x16 unshuffled input tile in LDS via async DMA
    // (global -> LDS without the VGPR round trip; tracked by ASYNCcnt).
    for (int i = tid; i < C_SQ_IN * 16; i += 256) {
        const int k = i >> 4, n = i & 15;
        const int grp = k >> 6, cin = k & 63;
        const int dy = grp & 1, dx = grp >> 1;   // concat order: (0,0),(1,0),(0,1),(1,1)
        const float* src =
            &x[((bi * 64 + cin) * 256 + (2 * y + dy)) * 256 + (2 * (x0 + n) + dx)];
        const unsigned lds_off = (unsigned)(unsigned long long)
            (__attribute__((address_space(3))) char*)&ldsX[i];
        async_g2lds_b32(lds_off, (unsigned long long)src);
    }
    wait_asynccnt0();        // this wave's DMAs are in LDS
    __syncthreads();         // ... and so are everyone else's

    const int wave = tid >> 5;
    const int l    = tid & 31;
    const int m0   = wave * 16;            // this wave's M tile
    const int mrow = l & 15;               // A: lanes hold rows M=0..15
    const int koff = (l >> 4) << 1;        // A/B: lane-half selects K pair {0,1} or {2,3}
    const int n    = l & 15;               // B/C/D: column across lanes

    v8f acc = {};
    #pragma unroll 4
    for (int k0 = 0; k0 < C_SQ_IN; k0 += 4) {
        v2f a, b;
        const float* wr = wsq + (m0 + mrow) * C_SQ_IN + k0 + koff;
        a.x = wr[0];
        a.y = wr[1];
        b.x = ldsX[(k0 + koff) * 16 + n];
        b.y = ldsX[(k0 + koff + 1) * 16 + n];
        acc = __builtin_amdgcn_wmma_f32_16x16x4_f32(
            /*neg_a=*/false, a, /*neg_b=*/false, b,
            /*c_mod=*/(short)0, acc, /*reuse_a=*/false, /*reuse_b=*/false);
    }

    const int hi8 = (l >> 4) * 8;          // C/D: lanes 16-31 hold M=v+8
    const int xg  = x0 + n;
    #pragma unroll
    for (int v = 0; v < 8; ++v) {
        const int m = m0 + v + hi8;        // squeeze output channel 0..127
        const float val = acc[v];
        if (m < 64) {
            up[((bi * 64 + m) * HW + y) * HW + xg] = val;
        } else {
            // low half: concat channel index == m
            out[((bi * 128 + shuffle_ch(m)) * HW + y) * HW + xg] = val;
        }
    }
}

// ---------------------------------------------------------------------------
// Kernel 2: fused gating (avgpool3 + dw-1x1 + softmax + dw-3x3 + BN/ReLU +
// product) into LDS, then the collapsed stride-3 conv as WMMA GEMM
// out[64] = w_conv[64 x 594] * g[594] per pixel. 4 waves x 16-row M tiles.
// ---------------------------------------------------------------------------
__global__ __launch_bounds__(128) void k_carafe(
    const float* __restrict__ up,      // (8,64,128,128) workspace
    const float* __restrict__ w_gw,    // (594)
    const float* __restrict__ w_gf,    // (594,9)
    const float* __restrict__ gamma_,  // (594)
    const float* __restrict__ beta_,   // (594)
    const float* __restrict__ mean_,   // (594)
    const float* __restrict__ var_,    // (594)
    const float* __restrict__ w_conv,  // (64,594)
    const float* __restrict__ b_conv,  // (64)
    float* __restrict__ out)           // (8,128,128,128)
{
    __shared__ float g[CKPAD * 16];    // 38,144 B

    const int p0  = blockIdx.x * 16;
    const int bi  = p0 >> 14;
    const int rem = p0 & 16383;
    const int y   = rem >> 7;
    const int x0  = rem & 127;
    const int tid = threadIdx.x;

    // zero the two K-padding rows (594,595)
    if (tid < 32) g[CK * 16 + tid] = 0.0f;

    // ---- Phase A: per-(channel, pixel) gating math -> g in LDS ----
    for (int task = tid; task < CIN_ * 16; task += 128) {
        const int c  = task >> 4;
        const int n  = task & 15;
        const int xc = x0 + n;

        float t[9];
        float s = 0.0f;
        #pragma unroll
        for (int dy = -1; dy <= 1; ++dy) {
            #pragma unroll
            for (int dx = -1; dx <= 1; ++dx) {
                const int yy = y + dy, xx = xc + dx;
                float v = 0.0f;
                if ((unsigned)yy < (unsigned)HW && (unsigned)xx < (unsigned)HW) {
                    if (c < 64)       v = up[((bi * 64 + c) * HW + yy) * HW + xx];
                    else if (c == 64) v = (float)xx * (2.0f / 127.0f) - 1.0f;  // xx coord
                    else              v = (float)yy * (2.0f / 127.0f) - 1.0f;  // yy coord
                }
                t[(dy + 1) * 3 + (dx + 1)] = v;
                s += v;
            }
        }
        const float avg = s * (1.0f / 9.0f);   // reference divides by 9 uniformly

        // softmax over the 9 gating logits (stable)
        float lg[9], mx = -3.4e38f;
        #pragma unroll
        for (int j = 0; j < 9; ++j) {
            lg[j] = avg * w_gw[c * 9 + j];
            mx = fmaxf(mx, lg[j]);
        }
        float den = 0.0f;
        #pragma unroll
        for (int j = 0; j < 9; ++j) { lg[j] = __expf(lg[j] - mx); den += lg[j]; }
        const float rden = 1.0f / den;

        #pragma unroll
        for (int j = 0; j < 9; ++j) {
            const int ch = c * 9 + j;
            float f = 0.0f;
            #pragma unroll
            for (int q = 0; q < 9; ++q) f += t[q] * w_gf[ch * 9 + q];
            const float sc = gamma_[ch] * rsqrtf(var_[ch] + 1e-5f);
            const float sh = beta_[ch] - mean_[ch] * sc;
            f = fmaxf(f * sc + sh, 0.0f);
            g[ch * 16 + n] = f * lg[j] * rden;
        }
    }
    __syncthreads();

    // ---- Phase B: GEMM out[64] = w_conv[64x594] * g[594] via WMMA ----
    const int wave = tid >> 5;
    const int l    = tid & 31;
    const int m0   = wave * 16;
    const int mrow = l & 15;
    const int koff = (l >> 4) << 1;
    const int n    = l & 15;

    v8f acc = {};
    // Main loop: k rows 0..591, no bounds checks (591 < 594).
    #pragma unroll 4
    for (int k0 = 0; k0 < CKPAD - 4; k0 += 4) {
        const int ka = k0 + koff;
        v2f a, b;
        const float* wr = w_conv + (m0 + mrow) * CK + ka;
        a.x = wr[0];
        a.y = wr[1];
        b.x = g[ka * 16 + n];
        b.y = g[(ka + 1) * 16 + n];
        acc = __builtin_amdgcn_wmma_f32_16x16x4_f32(
            false, a, false, b, (short)0, acc, false, false);
    }
    {   // Tail: k rows 592..595 (rows 594/595 are zero pads in g; guard A).
        const int ka = (CKPAD - 4) + koff;
        v2f a, b;
        a.x = (ka     < CK) ? w_conv[(m0 + mrow) * CK + ka]     : 0.0f;
        a.y = (ka + 1 < CK) ? w_conv[(m0 + mrow) * CK + ka + 1] : 0.0f;
        b.x = g[ka * 16 + n];
        b.y = g[(ka + 1) * 16 + n];
        acc = __builtin_amdgcn_wmma_f32_16x16x4_f32(
            false, a, false, b, (short)0, acc, false, false);
    }

    const int hi8 = (l >> 4) * 8;
    const int xg  = x0 + n;
    #pragma unroll
    for (int v = 0; v < 8; ++v) {
        const int o = m0 + v + hi8;               // up output channel 0..63
        const float val = acc[v] + b_conv[o];
        out[((bi * 128 + shuffle_ch(o)) * HW + y) * HW + xg] = val;
    }
}

extern "C" void kernel_launch(void* const* d_in, const int* in_sizes, int n_in,
                              void* d_out, int out_size, void* d_ws, size_t ws_size,
                              hipStream_t stream) {
    const float* x        = (const float*)d_in[0];
    const float* w_sq     = (const float*)d_in[1];
    const float* w_gw     = (const float*)d_in[2];
    const float* w_gf     = (const float*)d_in[3];
    const float* bn_gamma = (const float*)d_in[4];
    const float* bn_beta  = (const float*)d_in[5];
    const float* bn_mean  = (const float*)d_in[6];
    const float* bn_var   = (const float*)d_in[7];
    const float* w_conv   = (const float*)d_in[8];
    const float* b_conv   = (const float*)d_in[9];
    float* out = (float*)d_out;
    float* up  = (float*)d_ws;   // 33,554,432 bytes

    const int n_tiles = B_ * HW * (HW / 16);   // 8192 tiles of 16 pixels

    k_squeeze<<<n_tiles, 256, 0, stream>>>(x, w_sq, up, out);
    k_carafe<<<n_tiles, 128, 0, stream>>>(up, w_gw, w_gf, bn_gamma, bn_beta,
                                          bn_mean, bn_var, w_conv, b_conv, out);
}